// ViterbiDecoder_5901285065064
// MI455X (gfx1250) — compile-verified
//
#include <hip/hip_runtime.h>
#include <math.h>

// Viterbi (softmax semiring) forward DP, row-persistent tiled wavefront, gfx1250.
//
// Each cell publishes (m, E[3]) with m=max_s V[s], E=exp(V-m) so successors do
//   lse_k(A[s,k]+Vpred[k]) = m_pred + log(eA[s,:].E_pred)
// (3 exps + 3 logs + 9 FMA per cell instead of 9 exps + 3 logs).
//
// 2048x2048 cells -> 32 tile-rows of 64 rows. One persistent workgroup (64
// threads, 2 wave32) per tile-row sweeps 32 tiles of 64x64 left-to-right:
//  - left halo stays in LDS (same WG produced it) -> no Hcol, no left spin
//  - top halo via global Hrow + per-row progress counter (acquire/release)
//  - theta tiles streamed by the Tensor Data Mover, double-buffered in LDS:
//    issue load(tj+1), s_wait_tensorcnt(1) => load(tj) resident, DMA for tj+1
//    overlaps the 127-step internal wavefront.

#define BT 64                 // tile edge (threads per block)
#define NT 32                 // tiles per side (BT*NT == 2048)
#define NM 2048               // grid edge
#define TROW (BT * 3)         // theta dwords per LDS tile row (192)

typedef unsigned int u32;
typedef __attribute__((ext_vector_type(4))) u32 v4u;
typedef __attribute__((ext_vector_type(8))) int v8i;
typedef __attribute__((ext_vector_type(4))) int v4i;

__global__ void vit_init_flags(u32* flags, int n) {
    int t = blockIdx.x * blockDim.x + threadIdx.x;
    if (t < n) flags[t] = 0u;
}

// TDM 2D descriptor: 192 contiguous floats per row, 64 rows, row stride 6144
// floats, destination LDS. D# bit layout per CDNA5 ISA.
__device__ __forceinline__ void tdm_theta_load(const float* theta, int ti, int tj,
                                               float* dstLds) {
    unsigned long long gaddr = (unsigned long long)theta +
        ((unsigned long long)ti * BT * (NM * 3) + (unsigned long long)tj * BT * 3) * 4ull;
    u32 ldsoff = (u32)(unsigned long long)(void*)dstLds;   // low 32 bits = LDS byte offset

    v4u g0;
    g0[0] = 1u;                                   // count=1, user descriptor
    g0[1] = ldsoff;                               // lds_addr (bytes)
    g0[2] = (u32)gaddr;                           // global_addr[31:0]
    g0[3] = (u32)(gaddr >> 32) | (2u << 30);      // global_addr[56:32] | type=2

    v8i g1;
    g1[0] = (int)(2u << 16);                      // data_size=4B; no pad/iterate/mask
    g1[1] = (int)(((u32)(NM * 3) & 0xFFFFu) << 16);           // tensor_dim0 lo16 (6144)
    g1[2] = (int)((((u32)(NM * 3)) >> 16) | ((u32)NM << 16)); // dim0 hi | tensor_dim1 lo16
    g1[3] = (int)(((u32)NM >> 16) | ((u32)TROW << 16));       // dim1 hi | tile_dim0=192
    g1[4] = (int)(BT);                            // tile_dim1=64, tile_dim2=0
    g1[5] = (int)(NM * 3);                        // tensor_dim0_stride lo32 (6144)
    g1[6] = 0;
    g1[7] = 0;
    v4i gz4 = {0, 0, 0, 0};
    v8i gz8 = {0, 0, 0, 0, 0, 0, 0, 0};
    __builtin_amdgcn_tensor_load_to_lds(g0, g1, gz4, gz4, gz8, 0);
}

__global__ __launch_bounds__(BT) void vit_row_kernel(
    const float* __restrict__ theta, const float* __restrict__ A,
    float* __restrict__ out,
    float4* __restrict__ Hrow,   // [NT][NM]: (m,Ex,Em,Ey) of cell (tb*BT, j), j=1..NM
    u32* __restrict__ flags)     // [NT]: tiles completed by row-WG ti (monotonic)
{
    __shared__ float  thetaS[2][BT * TROW];   // 2 x 48 KB, TDM double buffer
    __shared__ float4 bufs[3][BT];            // rotating anti-diagonal (m,E) buffers
    __shared__ float4 topH[BT + 1];           // topH[k] = cell (ti*BT, tj*BT + k)
    __shared__ float4 leftH[2][BT];           // left halo ping-pong across tiles

    const int r  = threadIdx.x;
    const int ti = blockIdx.x;

    const float4 BZERO = make_float4(0.f, 1.f, 1.f, 1.f);   // V=(0,0,0)
    const float  EM1   = 0.36787944117144233f;              // e^-1

    // Transition weights in exp-space (wave-uniform).
    float eA[9];
#pragma unroll
    for (int q = 0; q < 9; ++q) eA[q] = __expf(A[q]);

    // Left halo for tj=0 is the j=0 boundary (V = 0).
    leftH[0][r] = BZERO;

    // Prime the TDM pipeline with tile tj=0 (wave 0 issues; EXEC-independent op).
    if (r < warpSize) tdm_theta_load(theta, ti, 0, &thetaS[0][0]);

#pragma unroll 1
    for (int tj = 0; tj < NT; ++tj) {
        const int lr  = tj & 1;        // LDS buffer / left-halo slot for this tile
        const int nlr = lr ^ 1;

        // Issue next tile's theta DMA into the other buffer (overlaps everything
        // below plus the whole compute phase of this tile).
        if (r < warpSize && tj + 1 < NT)
            tdm_theta_load(theta, ti, tj + 1, &thetaS[nlr][0]);

        // Wait for the row above to have produced our top halo (overlaps DMA).
        if (r == 0 && ti > 0) {
            while (__hip_atomic_load(&flags[ti - 1], __ATOMIC_ACQUIRE,
                                     __HIP_MEMORY_SCOPE_AGENT) < (u32)(tj + 1))
                __builtin_amdgcn_s_sleep(1);
        }

        // This tile's theta must be resident: in-order TDM => tensorcnt<=1
        // leaves only the just-issued load outstanding.
        if (r < warpSize) {
            if (tj + 1 < NT) __builtin_amdgcn_s_wait_tensorcnt(1);
            else             __builtin_amdgcn_s_wait_tensorcnt(0);
        }
        __syncthreads();

        // Stage top halo: topH[k] = cell (ti*BT, tj*BT + k), k=0..BT.
        for (int k = r; k <= BT; k += BT) {                  // thread 0 does k=0 and k=BT
            int j = tj * BT + k;
            float4 h;
            if (ti == 0) {
                h = (j == 0) ? make_float4(1.f, EM1, 1.f, EM1)   // V[0,0]=(0,1,0)
                             : BZERO;
            } else {
                h = (j == 0) ? BZERO : Hrow[(size_t)ti * NM + (j - 1)];
            }
            topH[k] = h;
        }
        __syncthreads();

        // 64x64 internal anti-diagonal wavefront.
        const float* thBase = &thetaS[lr][0];
        float4 left = leftH[lr][r];
        int cur = 0, prev = 2, prev2 = 1;
#pragma unroll 1
        for (int s = 0; s < 2 * BT - 1; ++s) {
            int c = s - r;
            if (0 <= c && c < BT) {
                float4 up, dg;
                if (r == 0) { up = topH[c + 1]; dg = topH[c]; }
                else        { up = bufs[prev][r - 1];
                              dg = (c == 0) ? leftH[lr][r - 1] : bufs[prev2][r - 1]; }
                float4 lf = left;

                const float* th = thBase + (r * TROW + 3 * c);
                float tx = th[0], tm = th[1], ty = th[2];

                float sx = eA[0] * up.y + eA[1] * up.z + eA[2] * up.w;
                float sm = eA[3] * dg.y + eA[4] * dg.z + eA[5] * dg.w;
                float sy = eA[6] * lf.y + eA[7] * lf.z + eA[8] * lf.w;
                float Vx = up.x + __logf(sx) + tx;
                float Vm = dg.x + __logf(sm) + tm;
                float Vy = lf.x + __logf(sy) + ty;

                float mn = fmaxf(Vx, fmaxf(Vm, Vy));
                float4 o;
                o.x = mn;
                o.y = __expf(Vx - mn);
                o.z = __expf(Vm - mn);
                o.w = __expf(Vy - mn);
                bufs[cur][r] = o;
                left = o;

                if (c == BT - 1) leftH[nlr][r] = o;          // next tile's left halo
                if (r == BT - 1 && ti + 1 < NT)              // row below's top halo
                    Hrow[(size_t)(ti + 1) * NM + (tj * BT + c)] = o;

                if (ti == NT - 1 && tj == NT - 1 && r == BT - 1 && c == BT - 1)
                    out[0] = mn + __logf(o.y + o.z + o.w);   // lse(V[N,M])
            }
            __syncthreads();
            int t = prev2; prev2 = prev; prev = cur; cur = t;
        }

        // Publish this tile: flush Hrow stores, then bump the progress counter.
        __threadfence();
        __syncthreads();
        if (r == 0)
            __hip_atomic_store(&flags[ti], (u32)(tj + 1), __ATOMIC_RELEASE,
                               __HIP_MEMORY_SCOPE_AGENT);
    }
}

extern "C" void kernel_launch(void* const* d_in, const int* in_sizes, int n_in,
                              void* d_out, int out_size, void* d_ws, size_t ws_size,
                              hipStream_t stream) {
    const float* theta = (const float*)d_in[0];   // (2048, 2048, 3) f32
    const float* A     = (const float*)d_in[1];   // (3, 3) f32
    float* out = (float*)d_out;                   // 1 f32

    // Workspace: Hrow (32*2048*16B = 1 MB) | flags (32*4B).
    char* ws = (char*)d_ws;
    float4* Hrow = (float4*)ws;
    u32*   flags = (u32*)(ws + (size_t)NT * NM * sizeof(float4));

    vit_init_flags<<<1, NT, 0, stream>>>(flags, NT);
    vit_row_kernel<<<NT, BT, 0, stream>>>(theta, A, out, Hrow, flags);
}